// LCNN_12893491823184
// MI455X (gfx1250) — compile-verified
//
#include <hip/hip_runtime.h>

#define LAT 16
#define NBATCH 2
#define NSITES (NBATCH * LAT * LAT * LAT * LAT) /* 131072 */
#define ND 4
#define NIN 10
#define NOUT 8
#define WCH (NIN * 9) /* 90 complex per site */
#define WAVES_PER_BLOCK 4

typedef __attribute__((ext_vector_type(2))) float v2f;
typedef __attribute__((ext_vector_type(8))) float v8f;

// Wave-private LDS producer/consumer sync: drain this wave's DS ops.
// (S tile is per-wave, region is divergence-free, so no s_barrier needed.)
__device__ __forceinline__ void lds_wave_sync() {
    asm volatile("s_wait_dscnt 0x0" ::: "memory");
}

// site = (((b*L + x0)*L + x1)*L + x2)*L + x3 ; lattice coord mu has shift (3-mu)*4
__device__ __forceinline__ int nbr_site(int s, int mu, int delta) {
    int sh = (3 - mu) * 4;
    int c  = (s >> sh) & (LAT - 1);
    int nc = (c + delta) & (LAT - 1);
    return s + ((nc - c) << sh);
}

__device__ __forceinline__ void load_mat(const float* __restrict__ Ure,
                                         const float* __restrict__ Uim,
                                         int site, int mu, float* mr, float* mi) {
    int base = (site * ND + mu) * 9;
#pragma unroll
    for (int e = 0; e < 9; ++e) { mr[e] = Ure[base + e]; mi[e] = Uim[base + e]; }
}

// C = A * B  (3x3 complex)
__device__ __forceinline__ void cmm(const float* ar, const float* ai,
                                    const float* br, const float* bi,
                                    float* cr, float* ci) {
#pragma unroll
    for (int a = 0; a < 3; ++a)
#pragma unroll
        for (int b = 0; b < 3; ++b) {
            float sr = 0.f, si = 0.f;
#pragma unroll
            for (int c = 0; c < 3; ++c) {
                float xr = ar[a*3+c], xi = ai[a*3+c];
                float yr = br[c*3+b], yi = bi[c*3+b];
                sr += xr * yr - xi * yi;
                si += xr * yi + xi * yr;
            }
            cr[a*3+b] = sr; ci[a*3+b] = si;
        }
}

// C = A * adj(B)  (3x3 complex)
__device__ __forceinline__ void cmm_adjB(const float* ar, const float* ai,
                                         const float* br, const float* bi,
                                         float* cr, float* ci) {
#pragma unroll
    for (int a = 0; a < 3; ++a)
#pragma unroll
        for (int b = 0; b < 3; ++b) {
            float sr = 0.f, si = 0.f;
#pragma unroll
            for (int d = 0; d < 3; ++d) {
                float xr = ar[a*3+d], xi = ai[a*3+d];
                float yr = br[b*3+d], yi = bi[b*3+d]; // conj
                sr += xr * yr + xi * yi;
                si += xi * yr - xr * yi;
            }
            cr[a*3+b] = sr; ci[a*3+b] = si;
        }
}

// W is stored interleaved-complex: Wc[(site*90 + j*9 + e)*2 + {re,im}]
__device__ __forceinline__ void store_w(float* __restrict__ Wc, int cidx,
                                        const float* pr, const float* pi) {
#pragma unroll
    for (int e = 0; e < 9; ++e) {
        v2f z; z.x = pr[e]; z.y = pi[e];
        *reinterpret_cast<v2f*>(Wc + 2 * (cidx + e)) = z;
    }
}

// ---- Kernel 1: plaquettes -> W channels 0..5 --------------------------------
__global__ void plaq_kernel(const float* __restrict__ Ure, const float* __restrict__ Uim,
                            float* __restrict__ Wc) {
    int tid = blockIdx.x * blockDim.x + threadIdx.x;
    if (tid >= NSITES * 6) return;
    int site = tid / 6, ch = tid % 6;
    const int MU[6] = {0, 0, 0, 1, 1, 2};
    const int NU[6] = {1, 2, 3, 2, 3, 3};
    int mu = MU[ch], nu = NU[ch];
    float m1r[9], m1i[9], m2r[9], m2i[9], m3r[9], m3i[9], m4r[9], m4i[9];
    load_mat(Ure, Uim, site, mu, m1r, m1i);                      // U_mu(x)
    load_mat(Ure, Uim, nbr_site(site, mu, 1), nu, m2r, m2i);     // U_nu(x+mu)
    load_mat(Ure, Uim, nbr_site(site, nu, 1), mu, m3r, m3i);     // U_mu(x+nu)
    load_mat(Ure, Uim, site, nu, m4r, m4i);                      // U_nu(x)
    float t1r[9], t1i[9], t2r[9], t2i[9], pr[9], pi[9];
    cmm(m1r, m1i, m2r, m2i, t1r, t1i);
    cmm_adjB(t1r, t1i, m3r, m3i, t2r, t2i);
    cmm_adjB(t2r, t2i, m4r, m4i, pr, pi);
    store_w(Wc, site * WCH + ch * 9, pr, pi);
}

// ---- Kernel 2: Polyakov loops -> W channels 6..9 ----------------------------
__global__ void poly_kernel(const float* __restrict__ Ure, const float* __restrict__ Uim,
                            float* __restrict__ Wc) {
    int tid = blockIdx.x * blockDim.x + threadIdx.x;
    if (tid >= NSITES * ND) return;
    int site = tid / ND, mu = tid % ND;
    float pr[9], pi[9], sr[9], si[9], tr[9], ti[9];
    load_mat(Ure, Uim, site, mu, pr, pi);
    for (int step = 1; step < LAT; ++step) {
        load_mat(Ure, Uim, nbr_site(site, mu, step), mu, sr, si);
        cmm(pr, pi, sr, si, tr, ti);
#pragma unroll
        for (int e = 0; e < 9; ++e) { pr[e] = tr[e]; pi[e] = ti[e]; }
    }
    store_w(Wc, site * WCH + (6 + mu) * 9, pr, pi);
}

// ---- Kernel 3: gauge-equivariant conv, j-contraction via V_WMMA_F32_16X16X4_F32
// Tile = 16 consecutive sites (a full t-line). Real GEMM: 16 rows (8 Re + 8 Im
// outputs) x K=60 (3 shifts x 10 j x {re,im}) x 16 columns (sites), looped over
// the 9 matrix elements e with double-buffered b64 B-fragment loads so the 15
// WMMAs per iteration run back-to-back behind one s_wait_loadcnt.
__global__ void __launch_bounds__(32 * WAVES_PER_BLOCK)
conv_kernel(const float* __restrict__ Ure, const float* __restrict__ Uim,
            const float* __restrict__ Wc,
            const float* __restrict__ Ore, const float* __restrict__ Oim,
            float* __restrict__ out) {
    __shared__ float S[WAVES_PER_BLOCK][16][NOUT][9][2];

    int lane = threadIdx.x & 31;
    int wave = threadIdx.x >> 5;
    int tile = blockIdx.x * WAVES_PER_BLOCK + wave;
    if (tile >= NSITES / 16) return;
    int site0 = tile * 16;
    int low = lane & 15;      // A: M row (0..15); B: column/site (0..15)
    int h   = lane >> 4;      // selects K sub-pair {0,1} vs {2,3} in each K-group
    int mysite = site0 + low;

    float acc[4][18];
#pragma unroll
    for (int t = 0; t < 4; ++t)
#pragma unroll
        for (int e = 0; e < 18; ++e) acc[t][e] = 0.f;

    for (int mu = 0; mu < ND; ++mu) {
        // --- A fragments (omega, independent of e): A[i,20k+2j+ri] real expansion
        v2f afrag[15];
        int irow = (low < 8) ? low : (low - 8);
#pragma unroll
        for (int g = 0; g < 15; ++g) {
            float vals[2];
#pragma unroll
            for (int p = 0; p < 2; ++p) {
                int E  = 4 * g + 2 * h + p;
                int kk = E / 20;
                int r  = E % 20;
                int j  = r >> 1, ri = r & 1;
                int oidx = ((irow * NIN + j) * ND + mu) * 3 + kk;
                float v;
                if (low < 8) v = ri ? -Oim[oidx] : Ore[oidx]; // Re-output rows
                else         v = ri ?  Ore[oidx] : Oim[oidx]; // Im-output rows
                vals[p] = v;
            }
            afrag[g].x = vals[0];
            afrag[g].y = vals[1];
        }

        // --- per-lane B base offsets (complex index): nbr*90 + j*9, float idx *2
        int boff[15];
        {
            int nbrs2[3];
#pragma unroll
            for (int kk = 0; kk < 3; ++kk)
                nbrs2[kk] = nbr_site(mysite, mu, kk - 1) * (WCH * 2);
#pragma unroll
            for (int g = 0; g < 15; ++g) {
                int E  = 4 * g + 2 * h;          // even -> (re,im) pair of one W elem
                int kk = E / 20;
                int j  = (E % 20) >> 1;
                boff[g] = nbrs2[kk] + j * 18;
            }
        }

        // --- e-loop with double-buffered B fragments (one b64 per K-group)
        v2f bb[2][15];
#pragma unroll
        for (int g = 0; g < 15; ++g)
            bb[0][g] = *reinterpret_cast<const v2f*>(Wc + boff[g]); // e = 0

        for (int e = 0; e < 9; ++e) {
            const int cur = e & 1, nxt = cur ^ 1;
            if (e < 8) {
#pragma unroll
                for (int g = 0; g < 15; ++g)
                    bb[nxt][g] = *reinterpret_cast<const v2f*>(
                        Wc + boff[g] + 2 * (e + 1));
            }
            v8f c = {0.f, 0.f, 0.f, 0.f, 0.f, 0.f, 0.f, 0.f};
#pragma unroll
            for (int g = 0; g < 15; ++g) {
                // D = A x B + C : 16x16x4 f32 WMMA
                c = __builtin_amdgcn_wmma_f32_16x16x4_f32(
                        false, afrag[g], false, bb[cur][g], (short)0, c,
                        false, false);
            }
            // C/D layout: lane<16 -> (M=v, N=lane) = Re rows; lane>=16 -> Im rows
#pragma unroll
            for (int v = 0; v < 8; ++v) S[wave][low][v][e][h] = c[v];
        }
        lds_wave_sync();

        // --- sandwich: W_conv[i] += A_mu S_i A_mu^dag ; 128 (site,i) tasks/wave
#pragma unroll
        for (int t = 0; t < 4; ++t) {
            int idx = lane + 32 * t;
            int sl = idx >> 3, ii = idx & 7;
            int sg = site0 + sl;
            float arr[9], aii[9];
            load_mat(Ure, Uim, sg, mu, arr, aii);
            float smr[9], smi[9];
#pragma unroll
            for (int e = 0; e < 9; ++e) {
                smr[e] = S[wave][sl][ii][e][0];
                smi[e] = S[wave][sl][ii][e][1];
            }
            float t1r[9], t1i[9], rr[9], rim[9];
            cmm(arr, aii, smr, smi, t1r, t1i);
            cmm_adjB(t1r, t1i, arr, aii, rr, rim);
#pragma unroll
            for (int e = 0; e < 9; ++e) {
                acc[t][2 * e]     += rr[e];
                acc[t][2 * e + 1] += rim[e];
            }
        }
        lds_wave_sync(); // reads drained before next mu overwrites S
    }

    // --- write interleaved complex64 output: ((site*8 + i)*9 + e)*2 + {re,im}
#pragma unroll
    for (int t = 0; t < 4; ++t) {
        int idx = lane + 32 * t;
        int sl = idx >> 3, ii = idx & 7;
        int sg = site0 + sl;
        int base = ((sg * NOUT + ii) * 9) * 2;
#pragma unroll
        for (int e = 0; e < 9; ++e) {
            v2f z; z.x = acc[t][2 * e]; z.y = acc[t][2 * e + 1];
            *reinterpret_cast<v2f*>(out + base + 2 * e) = z;
        }
    }
}

extern "C" void kernel_launch(void* const* d_in, const int* in_sizes, int n_in,
                              void* d_out, int out_size, void* d_ws, size_t ws_size,
                              hipStream_t stream) {
    (void)in_sizes; (void)n_in; (void)out_size; (void)ws_size;
    const float* Ure = (const float*)d_in[0];
    const float* Uim = (const float*)d_in[1];
    const float* Ore = (const float*)d_in[2];
    const float* Oim = (const float*)d_in[3];
    float* out = (float*)d_out;

    float* Wc = (float*)d_ws; // NSITES*90 interleaved complex (94.4 MB)

    {
        int total = NSITES * 6;
        plaq_kernel<<<(total + 255) / 256, 256, 0, stream>>>(Ure, Uim, Wc);
    }
    {
        int total = NSITES * ND;
        poly_kernel<<<(total + 255) / 256, 256, 0, stream>>>(Ure, Uim, Wc);
    }
    {
        int tiles = NSITES / 16; // 8192 waves of WMMA work
        int blocks = tiles / WAVES_PER_BLOCK;
        conv_kernel<<<blocks, 32 * WAVES_PER_BLOCK, 0, stream>>>(
            Ure, Uim, Wc, Ore, Oim, out);
    }
}